// KMeansGrouping_23819888624143
// MI455X (gfx1250) — compile-verified
//
#include <hip/hip_runtime.h>
#include <hip/hip_bf16.h>

typedef float v2f __attribute__((ext_vector_type(2)));
typedef float v8f __attribute__((ext_vector_type(8)));

constexpr int B = 32;
constexpr int N = 4096;
constexpr int D = 256;
constexpr int K = 24;
constexpr int KP = 32;          // clusters padded to 2 WMMA column tiles
constexpr int CS_STRIDE = 260;  // LDS center row stride (bank-conflict-free b64 reads)
constexpr int SC_STRIDE = 33;   // LDS score row stride

// ---------------------------------------------------------------- normalize
// inv[b*N+n] = 1 / max(||x_bn||, 1e-12); one wave per row, 8 rows per block.
__global__ __launch_bounds__(256) void norm_kernel(const float* __restrict__ X,
                                                   float* __restrict__ inv) {
  const int wave = threadIdx.x >> 5, lane = threadIdx.x & 31;
  const long row = (long)blockIdx.x * 8 + wave;
  const float* xr = X + row * D;
  float s = 0.f;
  for (int i = lane; i < D; i += 32) { float v = xr[i]; s += v * v; }
  for (int off = 16; off > 0; off >>= 1) s += __shfl_xor(s, off, 32);
  if (lane == 0) inv[row] = 1.0f / fmaxf(sqrtf(s), 1e-12f);
}

// ------------------------------------------------------------ init centers
__global__ __launch_bounds__(256) void init_centers_kernel(const float* __restrict__ X,
                                                           const float* __restrict__ inv,
                                                           float* __restrict__ centers) {
  const int gid = blockIdx.x * 256 + threadIdx.x;  // B*K*D threads
  const int d = gid & (D - 1);
  const int k = (gid >> 8) % K;
  const int b = gid / (K * D);
  const int idx = (int)((double)k * (double)(N - 1) / (double)(K - 1));
  centers[gid] = X[((long)b * N + idx) * D + d] * inv[b * N + idx];
}

// ----------------------------------------------------------- assignment GEMM
// scores = X_norm @ C^T via V_WMMA_F32_16X16X4_F32; argmin_k (c_sq - 2*score).
// grid = (N/128, B), block = 256 (8 waves, each wave owns a 16-row tile).
__global__ __launch_bounds__(256) void assign_kernel(const float* __restrict__ X,
                                                     const float* __restrict__ inv,
                                                     const float* __restrict__ centers,
                                                     int* __restrict__ assign_out) {
  __shared__ float cs[KP * CS_STRIDE];
  __shared__ float csq[KP];
  __shared__ float sc[8 * 16 * SC_STRIDE];

  const int tid = threadIdx.x;
  const int b = blockIdx.y;

  // Stage centers into LDS, zero-padded to 32 rows.
  for (int i = 0; i < (KP * D) / 256; ++i) {
    const int idx = i * 256 + tid;
    const int k = idx >> 8, d = idx & (D - 1);
    cs[k * CS_STRIDE + d] = (k < K) ? centers[((long)b * K + k) * D + d] : 0.0f;
  }
  __syncthreads();
  if (tid < KP) {
    float s = 0.f;
    if (tid < K)
      for (int d = 0; d < D; ++d) { float v = cs[tid * CS_STRIDE + d]; s += v * v; }
    csq[tid] = s;
  }
  __syncthreads();

  const int wave = tid >> 5, lane = tid & 31;
  const int hi = lane >> 4, lr = lane & 15;
  const int row = blockIdx.x * 128 + wave * 16 + lr;
  const float invv = inv[b * N + row];
  const float* xr = X + ((long)b * N + row) * D;
  const float* c0 = &cs[lr * CS_STRIDE];
  const float* c1 = &cs[(16 + lr) * CS_STRIDE];

  v8f acc0 = {}; v8f acc1 = {};
#pragma unroll 4
  for (int kk = 0; kk < D; kk += 4) {
    const int o = kk + 2 * hi;
    v2f a = *(const v2f*)(xr + o);
    a.x *= invv; a.y *= invv;
    const v2f b0 = *(const v2f*)(c0 + o);
    const v2f b1 = *(const v2f*)(c1 + o);
    acc0 = __builtin_amdgcn_wmma_f32_16x16x4_f32(false, a, false, b0, (short)0, acc0, false, false);
    acc1 = __builtin_amdgcn_wmma_f32_16x16x4_f32(false, a, false, b1, (short)0, acc1, false, false);
  }

  // Transpose per-wave scores through LDS: VGPR v holds row (v + 8*hi), col lr.
  float* scw = &sc[wave * 16 * SC_STRIDE];
#pragma unroll
  for (int v = 0; v < 8; ++v) {
    const int m = v + 8 * hi;
    scw[m * SC_STRIDE + lr] = acc0[v];
    scw[m * SC_STRIDE + 16 + lr] = acc1[v];
  }
  __syncthreads();

  if (lane < 16) {
    const float* sr = &scw[lr * SC_STRIDE];
    float best = 3.0e38f; int bi = 0;
    for (int k = 0; k < K; ++k) {
      const float v = csq[k] - 2.0f * sr[k];
      if (v < best) { best = v; bi = k; }
    }
    assign_out[b * N + blockIdx.x * 128 + wave * 16 + lr] = bi;
  }
}

// ------------------------------------------------------------ cluster update
// One workgroup per batch. Thread t exclusively owns dimension column t of the
// LDS accumulator, so the read-modify-write is race-free without atomics
// (deterministic). lc is updated only by thread 0.
__global__ __launch_bounds__(256) void update_kernel(const float* __restrict__ X,
                                                     const float* __restrict__ inv,
                                                     const int* __restrict__ assign_in,
                                                     float* __restrict__ sums,
                                                     float* __restrict__ counts) {
  __shared__ float ls[K * D];
  __shared__ float lc[K];
  const int tid = threadIdx.x, b = blockIdx.x;
  for (int i = tid; i < K * D; i += 256) ls[i] = 0.f;
  if (tid < K) lc[tid] = 0.f;
  __syncthreads();

  const float* xb = X + (long)b * N * D;
  const int* ab = assign_in + b * N;
  const float* ib = inv + b * N;
  for (int n = 0; n < N; ++n) {
    const int a = ab[n];
    const float x = xb[(long)n * D + tid] * ib[n];
    ls[a * D + tid] += x;
    if (tid == 0) lc[a] += 1.0f;
  }
  __syncthreads();
  for (int i = tid; i < K * D; i += 256) sums[(long)b * K * D + i] = ls[i];
  if (tid < K) counts[b * K + tid] = lc[tid];
}

// new_centers = counts>0 ? sums/counts : old centers
__global__ __launch_bounds__(256) void newcenters_kernel(const float* __restrict__ sums,
                                                         const float* __restrict__ counts,
                                                         float* __restrict__ centers) {
  const int gid = blockIdx.x * 256 + threadIdx.x;  // B*K*D
  const int k = (gid >> 8) % K;
  const int b = gid / (K * D);
  const float c = counts[b * K + k];
  if (c > 0.f) centers[gid] = sums[gid] / c;
}

// ---------------------------------------------------------------- finalize
__global__ __launch_bounds__(256) void out_centers_kernel(const float* __restrict__ centers,
                                                          float* __restrict__ out) {
  const int gid = blockIdx.x * 256 + threadIdx.x;  // B*K*D
  out[gid] = centers[gid];
}

__global__ __launch_bounds__(256) void zero_masks_kernel(float* __restrict__ outm) {
  const long gid = (long)blockIdx.x * 256 + threadIdx.x;  // B*K*N
  outm[gid] = 0.f;
}

__global__ __launch_bounds__(256) void scatter_masks_kernel(const int* __restrict__ assign_in,
                                                            float* __restrict__ outm) {
  const int gid = blockIdx.x * 256 + threadIdx.x;  // B*N
  const int b = gid / N, n = gid & (N - 1);
  const int a = assign_in[gid];
  outm[((long)b * K + a) * N + n] = 1.0f;
}

// ------------------------------------------------------------------ launcher
extern "C" void kernel_launch(void* const* d_in, const int* in_sizes, int n_in,
                              void* d_out, int out_size, void* d_ws, size_t ws_size,
                              hipStream_t stream) {
  (void)in_sizes; (void)n_in; (void)out_size; (void)ws_size;
  const float* X = (const float*)d_in[0];
  float* out = (float*)d_out;

  // Workspace layout (floats): inv | centers | sums | counts | assign(int)
  float* ws = (float*)d_ws;
  float* inv     = ws;                    // B*N      = 131072
  float* centers = inv + B * N;           // B*K*D    = 196608
  float* sums    = centers + B * K * D;   // B*K*D    = 196608
  float* counts  = sums + B * K * D;      // B*K      = 768
  int*   assign  = (int*)(counts + B * K);// B*N ints = 131072

  constexpr int N_ITERS = 25;

  norm_kernel<<<(B * N) / 8, 256, 0, stream>>>(X, inv);
  init_centers_kernel<<<(B * K * D) / 256, 256, 0, stream>>>(X, inv, centers);

  for (int it = 0; it < N_ITERS; ++it) {
    assign_kernel<<<dim3(N / 128, B), 256, 0, stream>>>(X, inv, centers, assign);
    update_kernel<<<B, 256, 0, stream>>>(X, inv, assign, sums, counts);
    newcenters_kernel<<<(B * K * D) / 256, 256, 0, stream>>>(sums, counts, centers);
  }
  // final assignment with converged centers
  assign_kernel<<<dim3(N / 128, B), 256, 0, stream>>>(X, inv, centers, assign);

  out_centers_kernel<<<(B * K * D) / 256, 256, 0, stream>>>(centers, out);
  float* outm = out + B * K * D;
  zero_masks_kernel<<<(B * K * N) / 256, 256, 0, stream>>>(outm);
  scatter_masks_kernel<<<(B * N) / 256, 256, 0, stream>>>(assign, outm);
}